// UnsupervisedWrapper_55035710931685
// MI455X (gfx1250) — compile-verified
//
#include <hip/hip_runtime.h>
#include <math.h>

typedef __attribute__((ext_vector_type(2))) float v2f;
typedef __attribute__((ext_vector_type(8))) float v8f;

#define N_PTS 8192
#define D_DIM 256
#define BM 64
#define BN 64
#define KC 64            // K chunk staged in LDS
#define LDSS 68          // padded LDS row stride (floats); 68 % 64 == 4 -> conflict-free b64 frags
#define TPB 128          // 4 waves of 32

struct Ws {
    float loss_sum;
    float reg_sum;
    float pad[2];
    unsigned long long keys[N_PTS];   // (ordered_f32 << 32) | (8191 - col)
};

// ---------------------------------------------------------------- helpers
__device__ __forceinline__ unsigned long long shfl_xor_u64(unsigned long long v, int m) {
    unsigned lo = (unsigned)v, hi = (unsigned)(v >> 32);
    lo = (unsigned)__shfl_xor((int)lo, m, 32);
    hi = (unsigned)__shfl_xor((int)hi, m, 32);
    return ((unsigned long long)hi << 32) | lo;
}

// stage a 64-row x 64-col fp32 chunk (rows row0..row0+63, cols kc..kc+63) into LDS
__device__ __forceinline__ void stage_chunk(const float* __restrict__ src, int row0, int kc,
                                            float* __restrict__ lds) {
    // 64 rows * 16 float4 = 1024 float4 chunks; 8 per thread
    for (int p = threadIdx.x; p < BM * (KC / 4); p += TPB) {
        int row = p >> 4;             // 16 float4 per row
        int c4  = (p & 15) << 2;      // float offset within chunk
        const float4 v = *reinterpret_cast<const float4*>(
            src + (size_t)(row0 + row) * D_DIM + kc + c4);
        *reinterpret_cast<float4*>(lds + row * LDSS + c4) = v;
    }
}

// one K-chunk of WMMA work: wave computes 16 rows x 64 cols into acc[0..3]
__device__ __forceinline__ void mma_chunk(const float* __restrict__ ldsA,
                                          const float* __restrict__ ldsB,
                                          int wave, int lane, v8f* acc) {
    const int hi = lane >> 4;        // 0: K=k0,k0+1  1: K=k0+2,k0+3 (per ISA layout)
    const int lo = lane & 15;
    const float* aBase = ldsA + (wave * 16 + lo) * LDSS + 2 * hi;
    for (int k0 = 0; k0 < KC; k0 += 4) {
        v2f a = *reinterpret_cast<const v2f*>(aBase + k0);
#pragma unroll
        for (int nt = 0; nt < 4; ++nt) {
            v2f b = *reinterpret_cast<const v2f*>(ldsB + (nt * 16 + lo) * LDSS + 2 * hi + k0);
            acc[nt] = __builtin_amdgcn_wmma_f32_16x16x4_f32(
                false, a, false, b, (short)0, acc[nt], false, false);
        }
    }
}

// ---------------------------------------------------------------- kernels
__global__ void init_ws_kernel(Ws* ws) {
    int i = blockIdx.x * blockDim.x + threadIdx.x;
    if (i == 0) { ws->loss_sum = 0.0f; ws->reg_sum = 0.0f; }
    if (i < N_PTS) ws->keys[i] = 0ull;
}

// loss: sum over all (i,j) of  (s < 1-1e-5 ? 1-s : 0) + (s > 0.5 ? s : 0), s = q[i].k[j]
__global__ void __launch_bounds__(TPB) loss_kernel(const float* __restrict__ q,
                                                   const float* __restrict__ k,
                                                   Ws* __restrict__ ws) {
    __shared__ alignas(16) float ldsQ[BM * LDSS];
    __shared__ alignas(16) float ldsK[BN * LDSS];
    const int r0 = blockIdx.y * BM;
    const int c0 = blockIdx.x * BN;
    const int lane = threadIdx.x & 31, wave = threadIdx.x >> 5;

    v8f acc[4] = {};
    for (int kc = 0; kc < D_DIM; kc += KC) {
        stage_chunk(q, r0, kc, ldsQ);
        stage_chunk(k, c0, kc, ldsK);
        __syncthreads();
        mma_chunk(ldsQ, ldsK, wave, lane, acc);
        __syncthreads();
    }

    float part = 0.0f;
#pragma unroll
    for (int nt = 0; nt < 4; ++nt)
#pragma unroll
        for (int e = 0; e < 8; ++e) {
            float s = acc[nt][e];
            part += (s < 1.0f - 1e-5f) ? (1.0f - s) : 0.0f;
            part += (s > 0.5f) ? s : 0.0f;
        }
    for (int m = 16; m >= 1; m >>= 1) part += __shfl_xor(part, m, 32);
    if (lane == 0) atomicAdd(&ws->loss_sum, part);
}

// nearest neighbor on q.qT with diagonal excluded: per-row atomicMax of packed key
__global__ void __launch_bounds__(TPB) nn_kernel(const float* __restrict__ q,
                                                 Ws* __restrict__ ws) {
    __shared__ alignas(16) float ldsA[BM * LDSS];
    __shared__ alignas(16) float ldsB[BN * LDSS];
    const int r0 = blockIdx.y * BM;
    const int c0 = blockIdx.x * BN;
    const int lane = threadIdx.x & 31, wave = threadIdx.x >> 5;
    const int lo = lane & 15, hi = lane >> 4;

    v8f acc[4] = {};
    for (int kc = 0; kc < D_DIM; kc += KC) {
        stage_chunk(q, r0, kc, ldsA);
        stage_chunk(q, c0, kc, ldsB);
        __syncthreads();
        mma_chunk(ldsA, ldsB, wave, lane, acc);
        __syncthreads();
    }

    // C layout: VGPR r, lanes 0-15 -> M=r, lanes 16-31 -> M=r+8; N = lane&15
#pragma unroll
    for (int r = 0; r < 8; ++r) {
        const int row = r0 + wave * 16 + r + 8 * hi;
        unsigned long long best = 0ull;
#pragma unroll
        for (int nt = 0; nt < 4; ++nt) {
            const int col = c0 + nt * 16 + lo;
            float s = acc[nt][r];
            unsigned u = __float_as_uint(s);
            u = (u & 0x80000000u) ? ~u : (u | 0x80000000u);   // order-preserving map
            unsigned long long key =
                ((unsigned long long)u << 32) | (unsigned)(8191 - col); // ties -> smaller col
            if (col == row) key = 0ull;                        // exclude diagonal
            if (key > best) best = key;
        }
        // reduce over the 16 columns held in this half-wave (masks stay within the half)
        for (int m = 1; m < 16; m <<= 1) {
            unsigned long long o = shfl_xor_u64(best, m);
            if (o > best) best = o;
        }
        if (lo == 0) atomicMax(&ws->keys[row], best);
    }
}

// regulariser: per row, d2 = sum((q[i]-q[nn]+1e-6)^2); accumulate log(d2)
__global__ void __launch_bounds__(256) reg_kernel(const float* __restrict__ q,
                                                  Ws* __restrict__ ws) {
    const int wave = threadIdx.x >> 5, lane = threadIdx.x & 31;
    const int row = blockIdx.x * 8 + wave;
    if (row >= N_PTS) return;
    const unsigned long long key = ws->keys[row];
    const int nn = 8191 - (int)(key & 0xFFFFFFFFull);
    const float* a = q + (size_t)row * D_DIM;
    const float* b = q + (size_t)nn * D_DIM;

    float d2 = 0.0f;
    const int c0 = lane * 8;                 // 32 lanes x 8 floats = 256
#pragma unroll
    for (int c = 0; c < 8; c += 4) {
        float4 va = *reinterpret_cast<const float4*>(a + c0 + c);
        float4 vb = *reinterpret_cast<const float4*>(b + c0 + c);
        float dx = va.x - vb.x + 1e-6f; d2 += dx * dx;
        float dy = va.y - vb.y + 1e-6f; d2 += dy * dy;
        float dz = va.z - vb.z + 1e-6f; d2 += dz * dz;
        float dw = va.w - vb.w + 1e-6f; d2 += dw * dw;
    }
    for (int m = 16; m >= 1; m >>= 1) d2 += __shfl_xor(d2, m, 32);
    if (lane == 0) atomicAdd(&ws->reg_sum, logf(d2));
}

__global__ void fin_kernel(const Ws* __restrict__ ws, float* __restrict__ out) {
    out[0] = ws->loss_sum / (float)N_PTS;                 // loss
    out[1] = -0.5f * ws->reg_sum / (float)N_PTS;          // reg = -mean(log(sqrt(d2)))
}

// ---------------------------------------------------------------- launch
extern "C" void kernel_launch(void* const* d_in, const int* in_sizes, int n_in,
                              void* d_out, int out_size, void* d_ws, size_t ws_size,
                              hipStream_t stream) {
    const float* q = (const float*)d_in[0];
    const float* k = (const float*)d_in[1];
    Ws* ws = (Ws*)d_ws;
    float* out = (float*)d_out;

    init_ws_kernel<<<(N_PTS + 255) / 256, 256, 0, stream>>>(ws);

    dim3 grid(N_PTS / BN, N_PTS / BM);
    loss_kernel<<<grid, TPB, 0, stream>>>(q, k, ws);
    nn_kernel<<<grid, TPB, 0, stream>>>(q, ws);
    reg_kernel<<<N_PTS / 8, 256, 0, stream>>>(q, ws);
    fin_kernel<<<1, 1, 0, stream>>>(ws, out);
}